// AnchorTargetLayer_64372969832515
// MI455X (gfx1250) — compile-verified
//
#include <hip/hip_runtime.h>
#include <hip/hip_bf16.h>
#include <math.h>

#define TPB      256
#define NB       2048      // histogram bins for exact rank selection
#define CAP      8192      // boundary-bin list capacity (expected ~60 entries)
#define MAXGT    256
#define MAXANC   128
#define RPN_NEG  0.3f
#define RPN_POS  0.7f
#define RPN_BATCH 256
#define RPN_NUMFG 128

typedef unsigned int u32x4 __attribute__((ext_vector_type(4)));
typedef int          i32x8 __attribute__((ext_vector_type(8)));
typedef int          i32x4 __attribute__((ext_vector_type(4)));

// ---------------------------------------------------------------- helpers ---

__device__ __forceinline__ float area_of(float x1, float y1, float x2, float y2) {
#pragma clang fp contract(off)
  return (x2 - x1 + 1.0f) * (y2 - y1 + 1.0f);
}

// IoU with FP contraction off so the column-max pass and the equality pass
// produce bit-identical values (is_best uses exact float equality).
__device__ __forceinline__ float iou_one(float ax1, float ay1, float ax2, float ay2, float aarea,
                                         float gx1, float gy1, float gx2, float gy2, float garea) {
#pragma clang fp contract(off)
  float iw    = fminf(ax2, gx2) - fmaxf(ax1, gx1) + 1.0f;
  float ih    = fminf(ay2, gy2) - fmaxf(ay1, gy1) + 1.0f;
  float inter = fmaxf(iw, 0.0f) * fmaxf(ih, 0.0f);
  return inter / (aarea + garea - inter);
}

__device__ __forceinline__ void anchor_of(int t, int A, int W, float fs, const float* s_anc,
                                          float& ax1, float& ay1, float& ax2, float& ay2) {
#pragma clang fp contract(off)
  int p = t / A, a = t - p * A;
  int gy = p / W, gx = p - gy * W;
  float sx = (float)gx * fs, sy = (float)gy * fs;
  ax1 = s_anc[a * 4 + 0] + sx;
  ay1 = s_anc[a * 4 + 1] + sy;
  ax2 = s_anc[a * 4 + 2] + sx;
  ay2 = s_anc[a * 4 + 3] + sy;
}

__device__ __forceinline__ int bin_of(float v) {
  int b = (int)(v * (float)NB);
  return b < 0 ? 0 : (b > NB - 1 ? NB - 1 : b);
}

// ------------------------------------------------ CDNA5 TDM staging to LDS --

#if __has_builtin(__builtin_amdgcn_tensor_load_to_lds)
__device__ __forceinline__ unsigned lds_addr_of(const void* p) {
  // LDS aperture: low 32 bits of a generic pointer are the LDS byte address.
  return (unsigned)(unsigned long long)p;
}

// Build a Tensor DMA descriptor (D#) for a 1-row tile of nElem f32 elements
// and issue TENSOR_LOAD_TO_LDS (CDNA5 ISA).
__device__ __forceinline__ void tdm_load_f32(const float* src, int nElem, void* lds_dst) {
  unsigned long long ga = (unsigned long long)src;
  u32x4 g0;
  g0.x = 1u;                                              // count=1, user mode, no gather
  g0.y = lds_addr_of(lds_dst);                            // lds_addr [63:32]
  g0.z = (unsigned)ga;                                    // global_addr [95:64]
  g0.w = ((unsigned)(ga >> 32) & 0x01FFFFFFu)             // global_addr [120:96]
       | (2u << 30);                                      // type=2 ("image") [127:126]
  i32x8 g1;
  g1[0] = 0x00020000;                                     // wg_mask=0 (not in cluster), data_size=4B
  g1[1] = (nElem & 0xFFFF) << 16;                         // tensor_dim0 low16 @bit48
  g1[2] = ((nElem >> 16) & 0xFFFF) | (1 << 16);           // tensor_dim0 hi | tensor_dim1=1 @bit80
  g1[3] = (nElem & 0xFFFF) << 16;                         // tile_dim0 [127:112]
  g1[4] = 1;                                              // tile_dim1=1, tile_dim2=0
  g1[5] = nElem;                                          // tensor_dim0_stride low32 @bit160
  g1[6] = (nElem & 0xFFFF) << 16;                         // tensor_dim1_stride low16 @bit208
  g1[7] = (nElem >> 16) & 0xFFFF;                         // tensor_dim1_stride hi
  i32x4 gz;
  gz[0] = 0; gz[1] = 0; gz[2] = 0; gz[3] = 0;             // groups 2/3: <=2D tensor
  i32x8 z8;
  z8[0] = 0; z8[1] = 0; z8[2] = 0; z8[3] = 0;
  z8[4] = 0; z8[5] = 0; z8[6] = 0; z8[7] = 0;
  __builtin_amdgcn_tensor_load_to_lds(g0, g1, gz, gz, z8, 0);
}
#endif

// Stage gt_boxes (M x 5 f32) into LDS + precompute gt areas. All threads must
// reach this (contains barriers).
__device__ __forceinline__ void stage_gt(const float* gt, int M, float* s_gt, float* s_ga) {
#if __has_builtin(__builtin_amdgcn_tensor_load_to_lds)
  if (threadIdx.x < 32) {                                 // wave 0 issues the DMA once
    tdm_load_f32(gt, M * 5, s_gt);
    __builtin_amdgcn_s_wait_tensorcnt(0);
  }
#else
  for (int i = threadIdx.x; i < M * 5; i += blockDim.x) s_gt[i] = gt[i];
#endif
  __syncthreads();
  for (int j = threadIdx.x; j < M; j += blockDim.x)
    s_ga[j] = area_of(s_gt[j * 5 + 0], s_gt[j * 5 + 1], s_gt[j * 5 + 2], s_gt[j * 5 + 3]);
  __syncthreads();
}

// ---------------------------------------------------------------- kernels ---

__global__ void k_init(unsigned* histFg, unsigned* histBg, int* sc) {
  int i = blockIdx.x * blockDim.x + threadIdx.x;
  int n = blockDim.x * gridDim.x;
  for (int k = i; k < NB; k += n) { histFg[k] = 0u; histBg[k] = 0u; }
  if (i < 64) sc[i] = 0;
}

// Pass 1: per-anchor row max / argmax / inside flag.
// temps into d_out: col0=inside, col5=max_ov, col6=argmax (float-exact, <128)
__global__ void k_rowpass(const float* __restrict__ gt, const float* __restrict__ anchors,
                          const float* __restrict__ im_info, const int* __restrict__ fsp,
                          int M, int A, int W, int T, float* __restrict__ out) {
  __shared__ float s_gt[MAXGT * 5];
  __shared__ float s_ga[MAXGT];
  __shared__ float s_anc[MAXANC];
  if (threadIdx.x < A * 4 && threadIdx.x < MAXANC) s_anc[threadIdx.x] = anchors[threadIdx.x];
  stage_gt(gt, M, s_gt, s_ga);

  int t = blockIdx.x * blockDim.x + threadIdx.x;
  if (t >= T) return;
  float fs  = (float)fsp[0];
  float imh = im_info[0], imw = im_info[1];
  float ax1, ay1, ax2, ay2;
  anchor_of(t, A, W, fs, s_anc, ax1, ay1, ax2, ay2);
  bool inside = (ax1 >= 0.0f) && (ay1 >= 0.0f) && (ax2 < imw) && (ay2 < imh);
  float aarea = area_of(ax1, ay1, ax2, ay2);

  float best = -2.0f; int bestj = 0;
#pragma unroll 4
  for (int j = 0; j < M; ++j) {
    float u = iou_one(ax1, ay1, ax2, ay2, aarea,
                      s_gt[j * 5 + 0], s_gt[j * 5 + 1], s_gt[j * 5 + 2], s_gt[j * 5 + 3], s_ga[j]);
    float v = inside ? u : -1.0f;
    if (v > best) { best = v; bestj = j; }   // strict '>' == jnp.argmax first-max
  }
  out[t * 7 + 0] = inside ? 1.0f : 0.0f;
  out[t * 7 + 5] = best;
  out[t * 7 + 6] = (float)bestj;
}

// Pass 2: per-GT column max over all T anchors (one block per GT, no atomics).
__global__ void k_colpass(const float* __restrict__ gt, const float* __restrict__ anchors,
                          const float* __restrict__ im_info, const int* __restrict__ fsp,
                          int A, int W, int T, float* __restrict__ gt_max) {
  __shared__ float s_anc[MAXANC];
  __shared__ float red[TPB];
  int j = blockIdx.x;
  if (threadIdx.x < A * 4 && threadIdx.x < MAXANC) s_anc[threadIdx.x] = anchors[threadIdx.x];
  __syncthreads();

  float gx1 = gt[j * 5 + 0], gy1 = gt[j * 5 + 1], gx2 = gt[j * 5 + 2], gy2 = gt[j * 5 + 3];
  float ga  = area_of(gx1, gy1, gx2, gy2);
  float fs  = (float)fsp[0];
  float imh = im_info[0], imw = im_info[1];

  float lmax = -1.0f;  // outside rows contribute -1; all v >= -1
  for (int t = threadIdx.x; t < T; t += blockDim.x) {
    float ax1, ay1, ax2, ay2;
    anchor_of(t, A, W, fs, s_anc, ax1, ay1, ax2, ay2);
    bool inside = (ax1 >= 0.0f) && (ay1 >= 0.0f) && (ax2 < imw) && (ay2 < imh);
    float aarea = area_of(ax1, ay1, ax2, ay2);
    float u = iou_one(ax1, ay1, ax2, ay2, aarea, gx1, gy1, gx2, gy2, ga);
    lmax = fmaxf(lmax, inside ? u : -1.0f);
  }
  red[threadIdx.x] = lmax;
  __syncthreads();
  for (int s = TPB / 2; s > 0; s >>= 1) {
    if (threadIdx.x < s) red[threadIdx.x] = fmaxf(red[threadIdx.x], red[threadIdx.x + s]);
    __syncthreads();
  }
  if (threadIdx.x == 0) gt_max[j] = red[0];
}

// Pass 3: is_best via exact equality vs column max, pre-subsample labels,
// regression targets. Writes d_out cols 0..4.
__global__ void k_labels(const float* __restrict__ gt, const float* __restrict__ anchors,
                         const float* __restrict__ im_info, const int* __restrict__ fsp,
                         const float* __restrict__ gt_max, int M, int A, int W, int T,
                         float* __restrict__ out) {
  __shared__ float s_gt[MAXGT * 5];
  __shared__ float s_ga[MAXGT];
  __shared__ float s_gm[MAXGT];
  __shared__ float s_anc[MAXANC];
  if (threadIdx.x < A * 4 && threadIdx.x < MAXANC) s_anc[threadIdx.x] = anchors[threadIdx.x];
  for (int j = threadIdx.x; j < M; j += blockDim.x) s_gm[j] = gt_max[j];
  stage_gt(gt, M, s_gt, s_ga);

  int t = blockIdx.x * blockDim.x + threadIdx.x;
  if (t >= T) return;
  float inside_f = out[t * 7 + 0];
  float maxov    = out[t * 7 + 5];
  int   bestj    = (int)out[t * 7 + 6];
  bool  inside   = inside_f != 0.0f;

  float fs = (float)fsp[0];
  float ax1, ay1, ax2, ay2;
  anchor_of(t, A, W, fs, s_anc, ax1, ay1, ax2, ay2);
  float aarea = area_of(ax1, ay1, ax2, ay2);

  bool isbest = false;
#pragma unroll 4
  for (int j = 0; j < M; ++j) {
    float u = iou_one(ax1, ay1, ax2, ay2, aarea,
                      s_gt[j * 5 + 0], s_gt[j * 5 + 1], s_gt[j * 5 + 2], s_gt[j * 5 + 3], s_ga[j]);
    float v = inside ? u : -1.0f;
    isbest = isbest || (v == s_gm[j]);
  }

  float lab = -1.0f;
  if (inside && maxov < RPN_NEG) lab = 0.0f;
  if (isbest)                    lab = 1.0f;
  if (inside && maxov >= RPN_POS) lab = 1.0f;

  float tx = 0.0f, ty = 0.0f, tw = 0.0f, th = 0.0f;
  if (inside) {
#pragma clang fp contract(off)
    float gx1 = s_gt[bestj * 5 + 0], gy1 = s_gt[bestj * 5 + 1];
    float gx2 = s_gt[bestj * 5 + 2], gy2 = s_gt[bestj * 5 + 3];
    float ew = ax2 - ax1 + 1.0f, eh = ay2 - ay1 + 1.0f;
    float ecx = ax1 + 0.5f * ew, ecy = ay1 + 0.5f * eh;
    float gw = gx2 - gx1 + 1.0f, gh = gy2 - gy1 + 1.0f;
    float gcx = gx1 + 0.5f * gw, gcy = gy1 + 0.5f * gh;
    tx = (gcx - ecx) / ew;
    ty = (gcy - ecy) / eh;
    tw = logf(gw / ew);
    th = logf(gh / eh);
  }
  out[t * 7 + 0] = lab;
  out[t * 7 + 1] = tx;
  out[t * 7 + 2] = ty;
  out[t * 7 + 3] = tw;
  out[t * 7 + 4] = th;
}

// Histograms of rand values over current fg / bg sets.
__global__ void k_hist(const float* __restrict__ out, const float* __restrict__ rfg,
                       const float* __restrict__ rbg, unsigned* histFg, unsigned* histBg, int T) {
  int t = blockIdx.x * blockDim.x + threadIdx.x;
  if (t >= T) return;
  float lab = out[t * 7];
  if (lab == 1.0f)       atomicAdd(&histFg[bin_of(rfg[t])], 1u);
  else if (lab == 0.0f)  atomicAdd(&histBg[bin_of(rbg[t])], 1u);
}

// Serial scan: find boundary bins for rank<K selection; num_bg & num_examples.
__global__ void k_scan(const unsigned* histFg, const unsigned* histBg, int* sc) {
  if (blockIdx.x != 0 || threadIdx.x != 0) return;
  int run = 0, binFg = NB, cumFg = 0;
  const int Kf = RPN_NUMFG;
  for (int b = 0; b < NB; ++b) {
    int h = (int)histFg[b];
    if (run < Kf && run + h > Kf) { binFg = b; cumFg = run; }
    run += h;
  }
  int nfg = run;
  int keptFg = nfg < Kf ? nfg : Kf;
  int Kb = RPN_BATCH - keptFg;
  run = 0; int binBg = NB, cumBg = 0;
  for (int b = 0; b < NB; ++b) {
    int h = (int)histBg[b];
    if (run < Kb && run + h > Kb) { binBg = b; cumBg = run; }
    run += h;
  }
  int nbg = run;
  int keptBg = nbg < Kb ? nbg : Kb;
  sc[0] = binFg; sc[1] = cumFg; sc[2] = Kf;
  sc[3] = binBg; sc[4] = cumBg; sc[5] = Kb;
  sc[6] = keptFg + keptBg;                    // num_examples (ranks are a permutation)
}

// Demote elements past the boundary bin; collect boundary-bin elements.
__global__ void k_collect(float* __restrict__ out, const float* __restrict__ rfg,
                          const float* __restrict__ rbg, const int* __restrict__ sc,
                          unsigned* fgIdx, float* fgVal, unsigned* bgIdx, float* bgVal,
                          int* cnt, int T) {
  int t = blockIdx.x * blockDim.x + threadIdx.x;
  if (t >= T) return;
  float lab = out[t * 7];
  if (lab == 1.0f) {
    float v = rfg[t]; int b = bin_of(v), bb = sc[0];
    if (b > bb) out[t * 7] = -1.0f;
    else if (b == bb) {
      int p = atomicAdd(&cnt[0], 1);
      if (p < CAP) { fgIdx[p] = (unsigned)t; fgVal[p] = v; }
    }
  } else if (lab == 0.0f) {
    float v = rbg[t]; int b = bin_of(v), bb = sc[3];
    if (b > bb) out[t * 7] = -1.0f;
    else if (b == bb) {
      int p = atomicAdd(&cnt[1], 1);
      if (p < CAP) { bgIdx[p] = (unsigned)t; bgVal[p] = v; }
    }
  }
}

// Exact within-bin ranks (value, then index tie-break == stable argsort).
__global__ void k_resolve(float* __restrict__ out, const int* __restrict__ sc,
                          const unsigned* fgIdx, const float* fgVal,
                          const unsigned* bgIdx, const float* bgVal, const int* cnt) {
  int nf = cnt[0] < CAP ? cnt[0] : CAP;
  int nb = cnt[1] < CAP ? cnt[1] : CAP;
  int cumFg = sc[1], Kf = sc[2], cumBg = sc[4], Kb = sc[5];
  for (int i = threadIdx.x; i < nf; i += blockDim.x) {
    float v = fgVal[i]; unsigned idx = fgIdx[i]; int r = cumFg;
    for (int j = 0; j < nf; ++j) {
      float u = fgVal[j];
      r += (u < v) || (u == v && fgIdx[j] < idx);
    }
    if (r >= Kf) out[(size_t)idx * 7] = -1.0f;
  }
  for (int i = threadIdx.x; i < nb; i += blockDim.x) {
    float v = bgVal[i]; unsigned idx = bgIdx[i]; int r = cumBg;
    for (int j = 0; j < nb; ++j) {
      float u = bgVal[j];
      r += (u < v) || (u == v && bgIdx[j] < idx);
    }
    if (r >= Kb) out[(size_t)idx * 7] = -1.0f;
  }
}

// Final weights.
__global__ void k_final(float* __restrict__ out, const int* __restrict__ sc, int T) {
  int t = blockIdx.x * blockDim.x + threadIdx.x;
  if (t >= T) return;
  float lab = out[t * 7];
  int ne = sc[6];
  float w = ne > 0 ? 1.0f / (float)ne : 0.0f;
  out[t * 7 + 5] = (lab == 1.0f) ? 1.0f : 0.0f;
  out[t * 7 + 6] = (lab >= 0.0f) ? w : 0.0f;
}

// ----------------------------------------------------------------- launch ---

extern "C" void kernel_launch(void* const* d_in, const int* in_sizes, int n_in,
                              void* d_out, int out_size, void* d_ws, size_t ws_size,
                              hipStream_t stream) {
  const float* gt    = (const float*)d_in[1];
  const float* iminf = (const float*)d_in[2];
  const float* anc   = (const float*)d_in[3];
  const float* rfg   = (const float*)d_in[4];
  const float* rbg   = (const float*)d_in[5];
  const int*   fsp   = (const int*)d_in[6];
  float* out = (float*)d_out;

  int M  = in_sizes[1] / 5;      // 128
  int A  = in_sizes[3] / 4;      // 9
  int T  = in_sizes[4];          // H*W*A
  int HW = T / A;
  int W  = (int)(sqrt((double)HW) + 0.5);  // square feature map (H == W)

  // workspace layout (~152 KB)
  float*    gtmax  = (float*)d_ws;
  unsigned* histFg = (unsigned*)((char*)d_ws + 4096);
  unsigned* histBg = histFg + NB;
  int*      sc     = (int*)(histBg + NB);
  int*      cnt    = sc + 32;
  unsigned* fgIdx  = (unsigned*)(sc + 64);
  float*    fgVal  = (float*)(fgIdx + CAP);
  unsigned* bgIdx  = (unsigned*)(fgVal + CAP);
  float*    bgVal  = (float*)(bgIdx + CAP);

  int grid = (T + TPB - 1) / TPB;
  k_init   <<<8, TPB, 0, stream>>>(histFg, histBg, sc);
  k_rowpass<<<grid, TPB, 0, stream>>>(gt, anc, iminf, fsp, M, A, W, T, out);
  k_colpass<<<M, TPB, 0, stream>>>(gt, anc, iminf, fsp, A, W, T, gtmax);
  k_labels <<<grid, TPB, 0, stream>>>(gt, anc, iminf, fsp, gtmax, M, A, W, T, out);
  k_hist   <<<grid, TPB, 0, stream>>>(out, rfg, rbg, histFg, histBg, T);
  k_scan   <<<1, 1, 0, stream>>>(histFg, histBg, sc);
  k_collect<<<grid, TPB, 0, stream>>>(out, rfg, rbg, sc, fgIdx, fgVal, bgIdx, bgVal, cnt, T);
  k_resolve<<<1, TPB, 0, stream>>>(out, sc, fgIdx, fgVal, bgIdx, bgVal, cnt);
  k_final  <<<grid, TPB, 0, stream>>>(out, sc, T);
}